// FieldProjector_4827543241370
// MI455X (gfx1250) — compile-verified
//
#include <hip/hip_runtime.h>

typedef float v2f __attribute__((ext_vector_type(2)));
typedef float v4f __attribute__((ext_vector_type(4)));
typedef float v8f __attribute__((ext_vector_type(8)));

#define NATOMS 512
#define NCHAN  4
#define KDIM   49
#define NDIM   8
#define NMESH  128
#define GPTS   1900                 // 50 radial * 38 angular
#define M_TOTAL (NATOMS*NCHAN)      // 2048
#define N_TOTAL (KDIM*NDIM)         // 392
#define M_TILES (M_TOTAL/32)        // 64  (32 rows per wave)
#define N_TILES ((N_TOTAL+15)/16)   // 25
#define KCHUNKS 237                 // 237*8 = 1896, + 4 tail

// ---------------------------------------------------------------------------
// Phase 1: quadratic-spline gather from the periodic mesh.
// Writes A[m=p*4+a][g] = interp[p,g,a] * weights[g]  (WMMA A operand, row-major K)
// ---------------------------------------------------------------------------
__global__ __launch_bounds__(256) void interp_kernel(
    const float* __restrict__ positions,   // [512][3]
    const float* __restrict__ mesh,        // [4][128][128][128]
    const float* __restrict__ cell,        // [3][3]
    const float* __restrict__ weights,     // [1900]
    const float* __restrict__ grid,        // [1900][3]
    float* __restrict__ A)                 // [2048][1900]
{
    const int tid = blockIdx.x * 256 + threadIdx.x;   // exactly 512*1900 threads
    const int p = tid / GPTS;
    const int g = tid - p * GPTS;

    // 3x3 inverse of cell (uniform -> scalar loads/ALU)
    const float c00=cell[0], c01=cell[1], c02=cell[2];
    const float c10=cell[3], c11=cell[4], c12=cell[5];
    const float c20=cell[6], c21=cell[7], c22=cell[8];
    const float det = c00*(c11*c22 - c12*c21)
                    - c01*(c10*c22 - c12*c20)
                    + c02*(c10*c21 - c11*c20);
    const float rd = 1.0f / det;
    const float i00 =  (c11*c22 - c12*c21)*rd;
    const float i01 = -(c01*c22 - c02*c21)*rd;
    const float i02 =  (c01*c12 - c02*c11)*rd;
    const float i10 = -(c10*c22 - c12*c20)*rd;
    const float i11 =  (c00*c22 - c02*c20)*rd;
    const float i12 = -(c00*c12 - c02*c10)*rd;
    const float i20 =  (c10*c21 - c11*c20)*rd;
    const float i21 = -(c00*c21 - c01*c20)*rd;
    const float i22 =  (c00*c11 - c01*c10)*rd;

    const float px = positions[p*3+0] + grid[g*3+0];
    const float py = positions[p*3+1] + grid[g*3+1];
    const float pz = positions[p*3+2] + grid[g*3+2];

    float uu[3];
    uu[0] = (px*i00 + py*i10 + pz*i20) * (float)NMESH;
    uu[1] = (px*i01 + py*i11 + pz*i21) * (float)NMESH;
    uu[2] = (px*i02 + py*i12 + pz*i22) * (float)NMESH;

    float w[3][3]; int ix[3][3];
    #pragma unroll
    for (int d = 0; d < 3; ++d) {
        const float fn = floorf(uu[d] + 0.5f);
        const float x  = uu[d] - fn;
        w[d][0] = 0.5f*(0.5f - x)*(0.5f - x);
        w[d][1] = 0.75f - x*x;
        w[d][2] = 0.5f*(0.5f + x)*(0.5f + x);
        const int ib = (int)fn;
        ix[d][0] = (ib - 1) & (NMESH - 1);   // periodic wrap, power-of-2 mesh
        ix[d][1] =  ib      & (NMESH - 1);
        ix[d][2] = (ib + 1) & (NMESH - 1);
    }

    int   off[27];
    float w3[27];
    int q = 0;
    #pragma unroll
    for (int i = 0; i < 3; ++i)
        #pragma unroll
        for (int j = 0; j < 3; ++j)
            #pragma unroll
            for (int k = 0; k < 3; ++k) {
                off[q] = (ix[0][i]*NMESH + ix[1][j])*NMESH + ix[2][k];
                w3[q]  = w[0][i]*w[1][j]*w[2][k];
                ++q;
            }

    const float wg = weights[g];
    #pragma unroll
    for (int a = 0; a < NCHAN; ++a) {
        const float* mp = mesh + (size_t)a * (NMESH*NMESH*NMESH);
        float s = 0.0f;
        #pragma unroll
        for (int o = 0; o < 27; ++o)
            s = fmaf(w3[o], mp[off[o]], s);
        A[(size_t)(p*NCHAN + a)*GPTS + g] = s * wg;
    }
}

// ---------------------------------------------------------------------------
// Phase 2: fp32 WMMA GEMM, 32x16 tile per wave, software-pipelined.
//   out[p,k,a,n] = sum_g A[p*4+a][g] * B[k*8+n][g]
// K is traversed in virtual chunks of 8: half-wave hl owns K = 8c+4*hl+{0..3},
// loaded as one b128 per lane per operand, feeding two WMMAs (.xy, .zw).
// This permutes the K order identically for A and B, which is legal since
// WMMA contracts over K.
// ---------------------------------------------------------------------------
__global__ __launch_bounds__(256) void gemm_kernel(
    const float* __restrict__ A,    // [2048][1900]
    const float* __restrict__ B,    // values: [49][8][1900] = [392][1900]
    float* __restrict__ out)        // [512][49][4][8]
{
    const int wid  = blockIdx.x * (256/32) + (threadIdx.x >> 5);  // 1600 waves
    const int lane = threadIdx.x & 31;
    const int tn   = wid % N_TILES;
    const int tm   = wid / N_TILES;        // 0..63
    const int hl   = lane >> 4;            // half-wave select
    const int id   = lane & 15;

    const int mrow0    = tm*32 + id;       // first M sub-tile row for this lane
    const int ncol_raw = tn*16 + id;
    const int ncol     = ncol_raw < N_TOTAL ? ncol_raw : (N_TOTAL - 1); // clamp: junk cols never stored

    const float* aptr0 = A + (size_t)mrow0        * GPTS + 4*hl;
    const float* aptr1 = A + (size_t)(mrow0 + 16) * GPTS + 4*hl;
    const float* bptr  = B + (size_t)ncol         * GPTS + 4*hl;

    v8f acc0 = {};
    v8f acc1 = {};

    // prologue: chunk 0
    v4f a0 = *(const v4f*)(aptr0);
    v4f a1 = *(const v4f*)(aptr1);
    v4f b  = *(const v4f*)(bptr);

    #pragma unroll 4
    for (int c = 1; c < KCHUNKS; ++c) {
        const v4f a0n = *(const v4f*)(aptr0 + 8*c);
        const v4f a1n = *(const v4f*)(aptr1 + 8*c);
        const v4f bn  = *(const v4f*)(bptr  + 8*c);

        const v2f blo = __builtin_shufflevector(b, b, 0, 1);
        const v2f bhi = __builtin_shufflevector(b, b, 2, 3);
        acc0 = __builtin_amdgcn_wmma_f32_16x16x4_f32(false, __builtin_shufflevector(a0, a0, 0, 1),
                                                     false, blo, (short)0, acc0, false, false);
        acc1 = __builtin_amdgcn_wmma_f32_16x16x4_f32(false, __builtin_shufflevector(a1, a1, 0, 1),
                                                     false, blo, (short)0, acc1, false, false);
        acc0 = __builtin_amdgcn_wmma_f32_16x16x4_f32(false, __builtin_shufflevector(a0, a0, 2, 3),
                                                     false, bhi, (short)0, acc0, false, false);
        acc1 = __builtin_amdgcn_wmma_f32_16x16x4_f32(false, __builtin_shufflevector(a1, a1, 2, 3),
                                                     false, bhi, (short)0, acc1, false, false);
        a0 = a0n; a1 = a1n; b = bn;
    }

    // epilogue WMMAs for the last full chunk
    {
        const v2f blo = __builtin_shufflevector(b, b, 0, 1);
        const v2f bhi = __builtin_shufflevector(b, b, 2, 3);
        acc0 = __builtin_amdgcn_wmma_f32_16x16x4_f32(false, __builtin_shufflevector(a0, a0, 0, 1),
                                                     false, blo, (short)0, acc0, false, false);
        acc1 = __builtin_amdgcn_wmma_f32_16x16x4_f32(false, __builtin_shufflevector(a1, a1, 0, 1),
                                                     false, blo, (short)0, acc1, false, false);
        acc0 = __builtin_amdgcn_wmma_f32_16x16x4_f32(false, __builtin_shufflevector(a0, a0, 2, 3),
                                                     false, bhi, (short)0, acc0, false, false);
        acc1 = __builtin_amdgcn_wmma_f32_16x16x4_f32(false, __builtin_shufflevector(a1, a1, 2, 3),
                                                     false, bhi, (short)0, acc1, false, false);
    }

    // K tail: 1896..1899, standard ISA chunk layout (K = 1896 + 2*hl + r)
    {
        const v2f at0 = *(const v2f*)(A + (size_t)mrow0        * GPTS + 1896 + 2*hl);
        const v2f at1 = *(const v2f*)(A + (size_t)(mrow0 + 16) * GPTS + 1896 + 2*hl);
        const v2f bt  = *(const v2f*)(B + (size_t)ncol         * GPTS + 1896 + 2*hl);
        acc0 = __builtin_amdgcn_wmma_f32_16x16x4_f32(false, at0, false, bt, (short)0, acc0, false, false);
        acc1 = __builtin_amdgcn_wmma_f32_16x16x4_f32(false, at1, false, bt, (short)0, acc1, false, false);
    }

    // D layout: c[r] -> M = r + 8*hl, N = id.  Scatter into out[p][k][a][n].
    const int n = tn*16 + id;
    if (n < N_TOTAL) {
        const int k  = n >> 3;          // / NDIM
        const int nn = n & 7;
        #pragma unroll
        for (int r = 0; r < 8; ++r) {
            const int m0 = tm*32 + r + 8*hl;        // acc0 rows
            const int m1 = m0 + 16;                  // acc1 rows
            const int p0 = m0 >> 2, ch0 = m0 & 3;
            const int p1 = m1 >> 2, ch1 = m1 & 3;
            out[(((size_t)p0*KDIM + k)*NCHAN + ch0)*NDIM + nn] = acc0[r];
            out[(((size_t)p1*KDIM + k)*NCHAN + ch1)*NDIM + nn] = acc1[r];
        }
    }
}

extern "C" void kernel_launch(void* const* d_in, const int* in_sizes, int n_in,
                              void* d_out, int out_size, void* d_ws, size_t ws_size,
                              hipStream_t stream) {
    const float* positions = (const float*)d_in[0];   // [512][3]
    const float* mesh      = (const float*)d_in[1];   // [4][128][128][128]
    const float* cell      = (const float*)d_in[2];   // [3][3]
    const float* values    = (const float*)d_in[3];   // [49][8][1900]
    const float* weights   = (const float*)d_in[4];   // [1900]
    const float* grid      = (const float*)d_in[5];   // [1900][3]
    float*       out       = (float*)d_out;           // [512][49][4][8]
    float*       A         = (float*)d_ws;            // [2048][1900] = 15.6 MB

    // Phase 1: 512*1900 = 972800 threads exactly (3800 blocks x 256)
    interp_kernel<<<(NATOMS*GPTS)/256, 256, 0, stream>>>(positions, mesh, cell,
                                                         weights, grid, A);

    // Phase 2: 64*25 = 1600 waves -> 200 blocks of 8 waves
    gemm_kernel<<<(M_TILES*N_TILES)/8, 256, 0, stream>>>(A, values, out);
}